// CTCLoss_57028575756840
// MI455X (gfx1250) — compile-verified
//
#include <hip/hip_runtime.h>
#include <cstddef>
#include <cstdint>

// CTC loss, MI455X (gfx1250).
// Dominant cost: one streaming read of pred (543 MB -> ~23us @ 23.3 TB/s).
// Strategy: per (b,t) row, async-copy the row into LDS once (CDNA5
// global_load_async_to_lds_b32), do max + sum-exp passes from LDS with
// float4 (ds_load_b128) reads, gather the 51 needed logits from LDS, then a
// tiny per-batch alpha recursion and the batch mean.

#define NEGV (-1e30f)

constexpr int Bb   = 128;
constexpr int Tt   = 160;
constexpr int Cc   = 6625;
constexpr int Ll   = 25;
constexpr int Ss   = 51;    // 2*L + 1
constexpr int LPS  = 52;    // padded row stride for lp_ext
constexpr int CPAD = 6656;  // 26 * 256 (row padded with NEGV); divisible by 4

// ---------------------------------------------------------------------------
// Kernel 1: per-(b,t) row logsumexp + gather of the S needed log-probs.
// One block per row; row staged into LDS via async global->LDS copies.
// ---------------------------------------------------------------------------
__global__ __launch_bounds__(256) void ctc_rowlse(const float* __restrict__ pred,
                                                  const int*   __restrict__ targets,
                                                  const int*   __restrict__ tlen,
                                                  float*       __restrict__ lp_ext) {
  __shared__ __align__(16) float row[CPAD];
  __shared__ float partial[8];
  __shared__ float sM, sLse;

  const int rid = blockIdx.x;           // b*T + t
  const int b   = rid / Tt;
  const int tid = threadIdx.x;
  const float* __restrict__ g = pred + (size_t)rid * Cc;

  // ---- Stage 6400 floats via CDNA5 async global->LDS (25 chunks x 1KB) ----
  // All 256 lanes active here (no divergence before this point): EXEC == all 1s.
  const unsigned lbase = (unsigned)(size_t)(&row[0]);  // low 32 bits = LDS byte addr
#pragma unroll 1
  for (int i = 0; i < 25; ++i) {
    unsigned           laddr = lbase + (unsigned)((i * 256 + tid) * 4);
    unsigned long long gaddr = (unsigned long long)(size_t)(g + i * 256 + tid);
    asm volatile("global_load_async_to_lds_b32 %0, %1, off"
                 :: "v"(laddr), "v"(gaddr)
                 : "memory");
  }
  // Tail 6400..6655: real data up to C-1, pad with NEGV (exp(NEGV-M) == 0).
  {
    const int idx = 6400 + tid;
    row[idx] = (idx < Cc) ? g[idx] : NEGV;
  }
  asm volatile("s_wait_asynccnt 0x0" ::: "memory");
  __syncthreads();

  const float4* __restrict__ row4 = reinterpret_cast<const float4*>(row);
  constexpr int N4 = CPAD / 4;  // 1664 float4 elements

  // ---- Pass 1: max over the row (ds_load_b128 + wave32 shuffle reduce) ----
  float m = NEGV;
#pragma unroll 2
  for (int i = tid; i < N4; i += 256) {
    const float4 v = row4[i];
    m = fmaxf(m, fmaxf(fmaxf(v.x, v.y), fmaxf(v.z, v.w)));
  }
  for (int off = 16; off; off >>= 1) m = fmaxf(m, __shfl_xor(m, off, 32));
  if ((tid & 31) == 0) partial[tid >> 5] = m;
  __syncthreads();
  if (tid < 32) {
    float w = (tid < 8) ? partial[tid] : NEGV;
    for (int off = 16; off; off >>= 1) w = fmaxf(w, __shfl_xor(w, off, 32));
    if (tid == 0) sM = w;
  }
  __syncthreads();
  const float M = sM;

  // ---- Pass 2: sum of exp(x - M) ----
  float s = 0.f;
#pragma unroll 2
  for (int i = tid; i < N4; i += 256) {
    const float4 v = row4[i];
    s += expf(v.x - M) + expf(v.y - M) + expf(v.z - M) + expf(v.w - M);
  }
  for (int off = 16; off; off >>= 1) s += __shfl_xor(s, off, 32);
  if ((tid & 31) == 0) partial[tid >> 5] = s;
  __syncthreads();
  if (tid < 32) {
    float w = (tid < 8) ? partial[tid] : 0.f;
    for (int off = 16; off; off >>= 1) w += __shfl_xor(w, off, 32);
    if (tid == 0) sLse = M + logf(w);
  }
  __syncthreads();
  const float lse = sLse;

  // ---- Gather: lp_ext[b,t,s] = pred[b,t,ext[s]] - lse, masked by validity ----
  if (tid < Ss) {
    const int len = tlen[b];
    const int cls = (tid & 1) ? targets[b * Ll + (tid >> 1)] : 0;  // blank == 0
    float lp = row[cls] - lse;
    if (tid >= 2 * len + 1) lp = NEGV;  // states beyond 2*len+1 masked
    lp_ext[(size_t)rid * LPS + tid] = lp;
  }
}

// ---------------------------------------------------------------------------
// Kernel 2: alpha recursion per batch element. One block (64 threads) per b.
// Replicates the reference's NEG arithmetic exactly.
// ---------------------------------------------------------------------------
__global__ __launch_bounds__(64) void ctc_alpha(const float* __restrict__ lp_ext,
                                                const int*   __restrict__ targets,
                                                const int*   __restrict__ tlen,
                                                float*       __restrict__ loss_b) {
  __shared__ float alpha[Ss];
  const int b = blockIdx.x;
  const int s = threadIdx.x;
  const float* __restrict__ lp = lp_ext + (size_t)b * Tt * LPS;
  const bool active = (s < Ss);

  // skip_ok[s]: s odd, s>=3, and tgt[s/2] != tgt[s/2 - 1]  (ext[s]!=blank auto)
  bool skip = false;
  if (active) {
    if ((s & 1) && s >= 3) {
      const int j = s >> 1;
      skip = (targets[b * Ll + j] != targets[b * Ll + j - 1]);
    }
    alpha[s] = (s == 0) ? lp[0] : (s == 1 ? lp[1] : NEGV);
  }
  __syncthreads();

  for (int t = 1; t < Tt; ++t) {
    float nv = 0.f;
    if (active) {
      const float a1 = alpha[s];
      const float a2 = (s >= 1) ? alpha[s - 1] : NEGV;
      const float a3 = (s >= 2 && skip) ? alpha[s - 2] : NEGV;
      float m = fmaxf(fmaxf(a1, a2), a3);
      m = fmaxf(m, NEGV);  // reference's NEG floor (m always finite)
      nv = lp[(size_t)t * LPS + s] +
           m + logf(expf(a1 - m) + expf(a2 - m) + expf(a3 - m));
    }
    __syncthreads();
    if (active) alpha[s] = nv;
    __syncthreads();
  }

  if (s == 0) {
    const int len = tlen[b];
    const float a1 = alpha[2 * len];
    const float a2 = alpha[2 * len - 1];
    float m = fmaxf(a1, a2);
    // jax.nn.logsumexp guards a non-finite max with 0:
    if (!(m > -1e37f)) m = 0.f;
    float l = -(m + logf(expf(a1 - m) + expf(a2 - m)));
    if (l >= 1e29f) l = 0.f;  // zero_infinity
    loss_b[b] = l / fmaxf((float)len, 1.f);
  }
}

// ---------------------------------------------------------------------------
// Kernel 3: mean over batch.
// ---------------------------------------------------------------------------
__global__ __launch_bounds__(128) void ctc_mean(const float* __restrict__ loss_b,
                                                float*       __restrict__ out) {
  __shared__ float wsum[4];
  float v = loss_b[threadIdx.x];
  for (int off = 16; off; off >>= 1) v += __shfl_xor(v, off, 32);
  if ((threadIdx.x & 31) == 0) wsum[threadIdx.x >> 5] = v;
  __syncthreads();
  if (threadIdx.x == 0)
    out[0] = (wsum[0] + wsum[1] + wsum[2] + wsum[3]) * (1.0f / (float)Bb);
}

// ---------------------------------------------------------------------------
extern "C" void kernel_launch(void* const* d_in, const int* in_sizes, int n_in,
                              void* d_out, int out_size, void* d_ws, size_t ws_size,
                              hipStream_t stream) {
  const float* pred    = (const float*)d_in[0];   // [B, T, C] fp32
  const int*   targets = (const int*)d_in[1];     // [B, L]
  const int*   tlen    = (const int*)d_in[2];     // [B]
  float*       out     = (float*)d_out;           // scalar

  float* lp_ext = (float*)d_ws;                         // B*T*LPS floats (~4.2 MB)
  float* loss_b = lp_ext + (size_t)Bb * Tt * LPS;       // B floats

  ctc_rowlse<<<Bb * Tt, 256, 0, stream>>>(pred, targets, tlen, lp_ext);
  ctc_alpha <<<Bb,       64, 0, stream>>>(lp_ext, targets, tlen, loss_b);
  ctc_mean  <<<1,       128, 0, stream>>>(loss_b, out);
}